// RLGate_56753697849547
// MI455X (gfx1250) — compile-verified
//
#include <hip/hip_runtime.h>
#include <math.h>

// ---- problem constants (from setup_inputs: B=4,T=2048,D=1024,E=8,K=2) ----
#define BT   8192          // B*T tokens
#define DDIM 1024          // model dim
#define NE   8             // experts
#define TOPK 2

// ---- gate kernel tiling ----
#define THREADS       128  // 4 waves (wave32) -> 128 blocks over 8192 tokens
#define TOK_PER_BLOCK 64   // 16 tokens per wave
#define CHUNK         32   // D-chunk staged in LDS per iteration
#define NCHUNK        (DDIM / CHUNK)
#define LDVEC         ((TOK_PER_BLOCK * CHUNK / 4) / THREADS) // float4 per thread per chunk
#define XS_STRIDE     36   // 32 + pad: (36*r + c) % 64 distinct over r=0..15, 16B-aligned rows
#define WT_STRIDE     1026 // 1024 + pad: (1026*n + c) % 64 = (2n+c)%64 distinct, 8B aligned

typedef float v2f __attribute__((ext_vector_type(2)));
typedef float v4f __attribute__((ext_vector_type(4)));
typedef float v8f __attribute__((ext_vector_type(8)));

// Gating: logits = x@W (+b), softmax, log-probs, Gumbel top-2, per-token
// REINFORCE contribution. One wave computes a 16(token)x16(expert-padded)
// logits tile via V_WMMA_F32_16X16X4_F32 accumulation over D.
// Double-buffered LDS staging overlaps the 32MB x stream with the WMMA chain.
__global__ __launch_bounds__(THREADS) void rlgate_gate(
    const float* __restrict__ x,        // [BT, D]
    const float* __restrict__ W,        // [D, E]
    const float* __restrict__ bvec,     // [E]
    const float* __restrict__ rewards,  // [BT]
    const float* __restrict__ baseline, // [1]
    const float* __restrict__ noise_u,  // [BT, E]
    int2*  __restrict__ idx_out,        // [BT] selected expert pair
    float* __restrict__ contrib)        // [BT] advantage * sum(selected logp)
{
    __shared__ float xs[2][TOK_PER_BLOCK * XS_STRIDE];       // 2 x 9 KB x tiles
    __shared__ float wt[(NE + 1) * WT_STRIDE];               // 36.1 KB W^T + zero row
    __shared__ float lt[THREADS / 32][16][NE];               //  2 KB logits tiles

    const int tid  = threadIdx.x;
    const int wave = tid >> 5;
    const int lane = tid & 31;
    const int half = lane >> 4;   // K-half selector per ISA A/B fragment layout
    const int m    = lane & 15;   // row (A) / col (B) index within fragment
    const int tb   = blockIdx.x * TOK_PER_BLOCK;

    // Stage W transposed into LDS once: wt[n][d] = W[d*NE + n] (coalesced reads),
    // plus one fully-zeroed pad row so padded-expert lanes load 0 with NO
    // divergent branch around the ds_load (keeps the WMMA loop clause-able).
    for (int f = tid; f < DDIM * NE; f += THREADS) {
        int d = f >> 3, n = f & 7;
        wt[n * WT_STRIDE + d] = W[f];
    }
    for (int f = tid; f < WT_STRIDE; f += THREADS)
        wt[NE * WT_STRIDE + f] = 0.0f;

    // Per-thread staging addresses (8 float4 per token row, 16 rows apart).
    size_t goff[LDVEC];
    int    soff[LDVEC];
    #pragma unroll
    for (int it = 0; it < LDVEC; ++it) {
        int v   = tid + it * THREADS;          // [0, TOK_PER_BLOCK*CHUNK/4)
        int row = v >> 3;                      // 8 float4 per row
        int cv  = v & 7;
        goff[it] = (size_t)(tb + row) * DDIM + cv * 4;
        soff[it] = row * XS_STRIDE + cv * 4;
    }

    // Preload chunk 0.
    v4f r[LDVEC];
    #pragma unroll
    for (int it = 0; it < LDVEC; ++it) r[it] = *(const v4f*)&x[goff[it]];
    #pragma unroll
    for (int it = 0; it < LDVEC; ++it) *(v4f*)&xs[0][soff[it]] = r[it];
    __syncthreads();

    v8f c = {};                                       // 16x16 f32 accumulator
    const int arow = (wave * 16 + m) * XS_STRIDE + 2 * half;
    const int brow = ((m < NE) ? m : NE) * WT_STRIDE + 2 * half; // zero row if padded

    for (int ci = 0; ci < NCHUNK; ++ci) {
        const int cur = ci & 1;
        // Prefetch next chunk into registers while WMMAs consume this one.
        if (ci + 1 < NCHUNK) {
            #pragma unroll
            for (int it = 0; it < LDVEC; ++it)
                r[it] = *(const v4f*)&x[goff[it] + (size_t)(ci + 1) * CHUNK];
        }
        // 8 x V_WMMA_F32_16X16X4_F32: A lane(m,half) = x[tok m][k+2h..+1],
        // B lane(n=m,half) = W^T[n][k+2h..+1] (zero row for n>=8). EXEC full.
        const float* xb = &xs[cur][0];
        #pragma unroll
        for (int ks = 0; ks < CHUNK / 4; ++ks) {
            v2f a  = *(const v2f*)&xb[arow + ks * 4];
            v2f bf = *(const v2f*)&wt[brow + ci * CHUNK + ks * 4];
            c = __builtin_amdgcn_wmma_f32_16x16x4_f32(
                    false, a, false, bf, (short)0, c, false, false);
        }
        if (ci + 1 < NCHUNK) {
            #pragma unroll
            for (int it = 0; it < LDVEC; ++it)
                *(v4f*)&xs[1 - cur][soff[it]] = r[it];
        }
        // Single barrier separates both hazard pairs:
        //  - this iter's stores to xs[1-cur] vs next iter's reads of it (RAW)
        //  - this iter's reads of xs[cur] vs next iter's stores to it (WAR)
        __syncthreads();
    }

    // Scatter logits tile: c[r] = logits[M = r + 8*half][N = m] (ISA C/D layout).
    if (m < NE) {
        #pragma unroll
        for (int rr = 0; rr < 8; ++rr)
            lt[wave][rr + 8 * half][m] = c[rr];
    }
    __syncthreads();

    // Per-token epilogue: 16 lanes of each wave each own one token.
    if (lane < 16) {
        const int tok = tb + wave * 16 + lane;
        float l[NE], mx = -INFINITY;
        #pragma unroll
        for (int e = 0; e < NE; ++e) {
            l[e] = lt[wave][lane][e] + bvec[e];
            mx = fmaxf(mx, l[e]);
        }
        float se = 0.0f;
        #pragma unroll
        for (int e = 0; e < NE; ++e) { l[e] = expf(l[e] - mx); se += l[e]; }
        const float inv = 1.0f / se;
        float lp[NE], key[NE];
        #pragma unroll
        for (int e = 0; e < NE; ++e) {
            lp[e] = logf(l[e] * inv + 1e-9f);
            float u = noise_u[(size_t)tok * NE + e] * (1.0f - 2e-7f) + 1e-7f;
            key[e] = lp[e] - logf(-logf(u));     // logp + Gumbel
        }
        // top-2 (descending, lowest index wins ties — matches lax.top_k)
        int i0 = 0; float k0 = key[0];
        #pragma unroll
        for (int e = 1; e < NE; ++e) if (key[e] > k0) { k0 = key[e]; i0 = e; }
        int i1 = (i0 == 0) ? 1 : 0; float k1 = key[i1];
        #pragma unroll
        for (int e = 0; e < NE; ++e)
            if (e != i0 && key[e] > k1) { k1 = key[e]; i1 = e; }

        idx_out[tok] = make_int2(i0, i1);
        float adv = rewards[tok] - baseline[0];
        contrib[tok] = adv * (lp[i0] + lp[i1]);
    }
}

// Combine: out[t,:] = 0.5*(eo[i0,t,:] + eo[i1,t,:]). Pure streaming, b128.
// Carries the roofline (96 MB of ~128 MB total). Single-use data -> NT hints
// keep the 192 MB L2 from being churned by this stream.
__global__ __launch_bounds__(256) void rlgate_combine(
    const float* __restrict__ eo,   // [NE, BT, D]
    const int2*  __restrict__ idx,  // [BT]
    float* __restrict__ out)        // [BT, D]
{
    const int tok = blockIdx.x;
    const int2 p  = idx[tok];
    const size_t ED   = (size_t)BT * DDIM;
    const size_t base = (size_t)tok * DDIM + threadIdx.x * 4;
    v4f a = __builtin_nontemporal_load((const v4f*)&eo[(size_t)p.x * ED + base]);
    v4f b = __builtin_nontemporal_load((const v4f*)&eo[(size_t)p.y * ED + base]);
    v4f rv = (a + b) * 0.5f;
    __builtin_nontemporal_store(rv, (v4f*)&out[base]);
}

// Deterministic fixed-order reduction (no float atomics -> replay-stable).
__global__ __launch_bounds__(256) void rlgate_reduce(
    const float* __restrict__ contrib, float* __restrict__ out_loss)
{
    __shared__ float sm[256];
    const int t = threadIdx.x;
    float s = 0.0f;
    for (int i = t; i < BT; i += 256) s += contrib[i];
    sm[t] = s;
    __syncthreads();
    for (int o = 128; o > 0; o >>= 1) {
        if (t < o) sm[t] += sm[t + o];
        __syncthreads();
    }
    if (t == 0) out_loss[0] = -sm[0] / (float)BT;
}

extern "C" void kernel_launch(void* const* d_in, const int* in_sizes, int n_in,
                              void* d_out, int out_size, void* d_ws, size_t ws_size,
                              hipStream_t stream) {
    const float* x    = (const float*)d_in[0]; // [B,T,D]
    const float* eo   = (const float*)d_in[1]; // [E,B,T,D]
    const float* rew  = (const float*)d_in[2]; // [B,T]
    const float* W    = (const float*)d_in[3]; // [D,E]
    const float* bv   = (const float*)d_in[4]; // [E]
    const float* base = (const float*)d_in[5]; // scalar
    const float* nu   = (const float*)d_in[6]; // [B,T,E]
    // d_in[7] = top_k == 2 (hardcoded as TOPK)

    float* out = (float*)d_out;                             // [BT*D] output + [1] aux loss
    int2*  idx = (int2*)d_ws;                               // 8192 * 8B  = 64 KB
    float* ctr = (float*)((char*)d_ws + BT * sizeof(int2)); // 8192 * 4B  = 32 KB

    rlgate_gate<<<BT / TOK_PER_BLOCK, THREADS, 0, stream>>>(
        x, W, bv, rew, base, nu, idx, ctr);
    rlgate_combine<<<BT, 256, 0, stream>>>(eo, idx, out);
    rlgate_reduce<<<1, 256, 0, stream>>>(ctr, out + (size_t)BT * DDIM);
}